// NestedGIN_38474317038530
// MI455X (gfx1250) — compile-verified
//
#include <hip/hip_runtime.h>

// NestedGIN on MI455X (gfx1250, wave32).
// Phases:
//   1) layer1: scalar edge scatter (x is [N,1]) -> outer-product expand -> WMMA GEMM
//   2) layers 2,3: float4 edge scatter-add (L2-resident) -> fused (h+agg)@W1 GEMM -> @W2 GEMM
//   3) two-stage segment-sum pooling N -> S -> G
// GEMMs use V_WMMA_F32_16X16X4_F32 to stay bit-faithful fp32.

typedef float v2f __attribute__((ext_vector_type(2)));
typedef float v8f __attribute__((ext_vector_type(8)));

#define HDIM 128
#define LDS_PITCH 132  // 16x128 A tile padded: bank = (4*row + k) % 64, conflict-free

// ---------------- layer 1 (features are scalar per node) ----------------

__global__ void edge_agg_scalar(const int* __restrict__ ei, const float* __restrict__ x,
                                float* __restrict__ agg1, int E) {
    int e = blockIdx.x * blockDim.x + threadIdx.x;
    if (e >= E) return;
    int s = ei[e];
    int d = ei[E + e];
    atomicAdd(&agg1[d], x[s]);
}

__global__ void layer1_expand(const float* __restrict__ x, const float* __restrict__ agg1,
                              const float* __restrict__ W1, const float* __restrict__ b1,
                              float* __restrict__ t, int N) {
    int tid = blockIdx.x * blockDim.x + threadIdx.x;
    int i = tid >> 7;        // node
    int j = tid & 127;       // feature
    if (i >= N) return;
    float s = x[i] + agg1[i];
    float v = s * W1[j] + b1[j];
    t[tid] = v > 0.0f ? v : 0.0f;
}

// ---------------- vector scatter-add: one wave per edge ----------------

__global__ void edge_agg_vec(const int* __restrict__ ei, const float* __restrict__ h,
                             float* __restrict__ agg, int E) {
    int tid = blockIdx.x * blockDim.x + threadIdx.x;
    int e = tid >> 5;              // edge
    int c = (tid & 31) << 2;       // feature chunk of 4
    if (e >= E) return;
    int s = ei[e];
    int d = ei[E + e];
    float4 v = *(const float4*)(h + (long)s * HDIM + c);
    float* p = agg + (long)d * HDIM + c;
    atomicAdd(p + 0, v.x);
    atomicAdd(p + 1, v.y);
    atomicAdd(p + 2, v.z);
    atomicAdd(p + 3, v.w);
}

// ---------------- fp32 WMMA GEMM: out = relu((A [+ A2]) @ W + bias) ----------------
// A: [M,128], W: [128,128] row-major, out: [M,128]
// Block = 256 threads (8 waves) = one 16-row tile; wave w owns columns [16w, 16w+16).

__global__ __launch_bounds__(256)
void gemm_relu_wmma(const float* __restrict__ A, const float* __restrict__ A2, int has2,
                    const float* __restrict__ W, const float* __restrict__ bias,
                    float* __restrict__ out, int M) {
    __shared__ float lds_a[16 * LDS_PITCH];

    int m0 = blockIdx.x * 16;

    // Stage 16x128 A tile (with fused elementwise add) into LDS: 512 float4s / 256 threads.
    for (int v = 0; v < 2; ++v) {
        int idx4 = threadIdx.x * 2 + v;      // 0..511
        int row  = idx4 >> 5;                // 32 float4 per 128-float row
        int col  = (idx4 & 31) << 2;
        float4 av = make_float4(0.f, 0.f, 0.f, 0.f);
        if (m0 + row < M) {
            long gi = (long)(m0 + row) * HDIM + col;
            av = *(const float4*)(A + gi);
            if (has2) {
                float4 bv = *(const float4*)(A2 + gi);
                av.x += bv.x; av.y += bv.y; av.z += bv.z; av.w += bv.w;
            }
        }
        *(float4*)(lds_a + row * LDS_PITCH + col) = av;
    }
    __syncthreads();

    int lane = threadIdx.x & 31;
    int wv   = threadIdx.x >> 5;
    int half = lane >> 4;      // 0: lower 16 lanes (K pair 0,1), 1: upper (K pair 2,3)
    int lh   = lane & 15;
    int c0   = wv * 16;

    v8f acc = {0.f, 0.f, 0.f, 0.f, 0.f, 0.f, 0.f, 0.f};

    const float* Wp = W + c0 + lh;   // column base for this lane's B fragment

    for (int k = 0; k < HDIM; k += 4) {
        // A fragment (ISA 16x4 f32 layout): lanes 0-15 -> A[lh][k+0..1], lanes 16-31 -> A[lh][k+2..3]
        float2 av = *(const float2*)(lds_a + lh * LDS_PITCH + k + 2 * half);
        v2f a; a.x = av.x; a.y = av.y;
        // B fragment (4x16): v0 = W[k+2h][c], v1 = W[k+2h+1][c]
        v2f b;
        b.x = Wp[(k + 2 * half)     * HDIM];
        b.y = Wp[(k + 2 * half + 1) * HDIM];
        acc = __builtin_amdgcn_wmma_f32_16x16x4_f32(
            /*neg_a=*/false, a, /*neg_b=*/false, b,
            /*c_mod=*/(short)0, acc, /*reuse_a=*/false, /*reuse_b=*/false);
    }

    // C/D layout: VGPR r -> row r (lanes 0-15) / row r+8 (lanes 16-31), col = lane%16
    float bb = bias[c0 + lh];
    #pragma unroll
    for (int r = 0; r < 8; ++r) {
        int row = m0 + r + 8 * half;
        if (row < M) {
            float v = acc[r] + bb;
            out[(long)row * HDIM + c0 + lh] = v > 0.0f ? v : 0.0f;
        }
    }
}

// ---------------- pooling ----------------

__global__ void pool_nodes(const float* __restrict__ h, const int* __restrict__ n2s,
                           float* __restrict__ subg, int N) {
    int tid = blockIdx.x * blockDim.x + threadIdx.x;
    int i = tid >> 5;
    int c = (tid & 31) << 2;
    if (i >= N) return;
    int sg = n2s[i];
    float4 v = *(const float4*)(h + (long)i * HDIM + c);
    float* p = subg + (long)sg * HDIM + c;
    atomicAdd(p + 0, v.x);
    atomicAdd(p + 1, v.y);
    atomicAdd(p + 2, v.z);
    atomicAdd(p + 3, v.w);
}

__global__ void pool_sub(const float* __restrict__ subg, const int* __restrict__ s2g,
                         float* __restrict__ out, int S) {
    int tid = blockIdx.x * blockDim.x + threadIdx.x;
    int s = tid >> 5;
    int c = (tid & 31) << 2;
    if (s >= S) return;
    int g = s2g[s];
    float4 v = *(const float4*)(subg + (long)s * HDIM + c);
    float* p = out + (long)g * HDIM + c;
    atomicAdd(p + 0, v.x);
    atomicAdd(p + 1, v.y);
    atomicAdd(p + 2, v.z);
    atomicAdd(p + 3, v.w);
}

// ---------------- driver ----------------

extern "C" void kernel_launch(void* const* d_in, const int* in_sizes, int n_in,
                              void* d_out, int out_size, void* d_ws, size_t ws_size,
                              hipStream_t stream) {
    const float* x    = (const float*)d_in[0];
    const int*   ei   = (const int*)d_in[1];
    const int*   n2s  = (const int*)d_in[2];
    const int*   s2g  = (const int*)d_in[3];
    const float* c1W1 = (const float*)d_in[4];
    const float* c1b1 = (const float*)d_in[5];
    const float* c1W2 = (const float*)d_in[6];
    const float* c1b2 = (const float*)d_in[7];
    const float* cW1  = (const float*)d_in[8];
    const float* cb1  = (const float*)d_in[9];
    const float* cW2  = (const float*)d_in[10];
    const float* cb2  = (const float*)d_in[11];

    const int N = in_sizes[0];       // x is [N,1]
    const int E = in_sizes[1] / 2;   // edge_index [2,E]
    const int S = in_sizes[3];       // subgraph_to_graph [S]
    const size_t NH = (size_t)N * HDIM;

    float* out    = (float*)d_out;
    float* hbuf   = (float*)d_ws;            // [N,128]
    float* tbuf   = hbuf + NH;               // [N,128]
    float* aggbuf = tbuf + NH;               // [N,128]
    float* agg1   = aggbuf + NH;             // [N]
    float* subg   = agg1 + N;                // [S,128]

    const int MT = (N + 15) / 16;            // 6250 row tiles

    // ---- layer 1 ----
    hipMemsetAsync(agg1, 0, (size_t)N * sizeof(float), stream);
    edge_agg_scalar<<<(E + 255) / 256, 256, 0, stream>>>(ei, x, agg1, E);
    layer1_expand<<<(int)((NH + 255) / 256), 256, 0, stream>>>(x, agg1, c1W1, c1b1, tbuf, N);
    gemm_relu_wmma<<<MT, 256, 0, stream>>>(tbuf, tbuf, 0, c1W2, c1b2, hbuf, N);

    // ---- layers 2, 3 ----
    for (int L = 0; L < 2; ++L) {
        hipMemsetAsync(aggbuf, 0, NH * sizeof(float), stream);
        edge_agg_vec<<<(int)(((size_t)E * 32 + 255) / 256), 256, 0, stream>>>(ei, hbuf, aggbuf, E);
        gemm_relu_wmma<<<MT, 256, 0, stream>>>(hbuf, aggbuf, 1,
                                               cW1 + (size_t)L * HDIM * HDIM, cb1 + L * HDIM,
                                               tbuf, N);
        gemm_relu_wmma<<<MT, 256, 0, stream>>>(tbuf, tbuf, 0,
                                               cW2 + (size_t)L * HDIM * HDIM, cb2 + L * HDIM,
                                               hbuf, N);
    }

    // ---- pooling N -> S -> G ----
    hipMemsetAsync(subg, 0, (size_t)S * HDIM * sizeof(float), stream);
    hipMemsetAsync(out, 0, (size_t)out_size * sizeof(float), stream);
    pool_nodes<<<(int)(((size_t)N * 32 + 255) / 256), 256, 0, stream>>>(hbuf, n2s, subg, N);
    pool_sub<<<(int)(((size_t)S * 32 + 255) / 256), 256, 0, stream>>>(subg, s2g, out, S);
}